// Quantizer_44753559225057
// MI455X (gfx1250) — compile-verified
//
#include <hip/hip_runtime.h>
#include <math.h>

typedef __attribute__((ext_vector_type(2))) float v2f;
typedef __attribute__((ext_vector_type(8))) float v8f;

// Problem shape (fixed by reference)
#define BATCH  64
#define CDIM   128
#define EDIM   64
#define KCODES 512
#define HWPIX  1024   // 32*32
#define TILE   16
#define NPIX   (BATCH * HWPIX)   // 65536
#define NTILES (NPIX / TILE)     // 4096

// ---------------------------------------------------------------------------
// Kernel 0: codebook squared norms -> ws[0..511]
// ---------------------------------------------------------------------------
__global__ __launch_bounds__(512) void vq_prep_kernel(const float* __restrict__ embed_w,
                                                      float* __restrict__ codenorm) {
    int k = threadIdx.x;
    if (k < KCODES) {
        const float* ev = embed_w + (size_t)k * EDIM;
        float s = 0.f;
#pragma unroll 8
        for (int e = 0; e < EDIM; ++e) { float v = ev[e]; s += v * v; }
        codenorm[k] = s;
    }
}

// ---------------------------------------------------------------------------
// Kernel 1: fused projection + distance GEMM + argmax + log_softmax + gather.
// One block = 16 pixels (one image-contiguous strip), 128 threads = 4 waves.
// ---------------------------------------------------------------------------
__global__ __launch_bounds__(128) void vq_main_kernel(
    const float* __restrict__ z,        // [B,C,H,W]
    const float* __restrict__ proj_w,   // [E,C]
    const float* __restrict__ proj_b,   // [E]
    const float* __restrict__ embed_w,  // [K,E]
    const float* __restrict__ codenorm, // [K]
    float* __restrict__ out_zq,         // [B,E,H,W]
    float* __restrict__ out_ind,        // [B,H,W] (as float)
    float* __restrict__ out_lp,         // [B,K,H,W]
    float* __restrict__ diff_part)      // [NTILES]
{
    __shared__ float zel[TILE][EDIM + 4];     // z_e tile, padded row stride
    __shared__ float xs[TILE][KCODES + 1];    // -dist tile (16 x 512)
    __shared__ float rownorm[TILE];
    __shared__ float rowmax[TILE];
    __shared__ float rowlse[TILE];
    __shared__ int   rowbest[TILE];
    __shared__ float pval[128];
    __shared__ int   pidx[128];

    const int t     = threadIdx.x;
    const int wave  = t >> 5;
    const int lane  = t & 31;
    const int col   = lane & 15;       // fragment column (N) / A row (M)
    const int hi    = lane >> 4;
    const int kbase = hi * 2;          // K-subindex base within a fragment
    const int wg    = blockIdx.x;
    const int b     = wg >> 6;         // 64 tiles per image
    const int hw0   = (wg & 63) * TILE;
    const int n0    = wg * TILE;

    // ---- GEMM1: z_e[16 x 64] = z_tile[16 x 128] * proj_w^T + b   (f32 WMMA)
    {
        const int e0 = wave * 16;                       // wave's 16 embedding cols
        const float* zbase = z + (size_t)b * CDIM * HWPIX + hw0;
        v8f acc = {};
#pragma unroll
        for (int s = 0; s < CDIM / 4; ++s) {
            const int c = s * 4 + kbase;
            v2f a;
            a.x = zbase[(size_t)c * HWPIX + col];       // A[M=col, K=c]
            a.y = zbase[(size_t)(c + 1) * HWPIX + col];
            const float* pw = proj_w + (size_t)(e0 + col) * CDIM + c;
            v2f bm; bm.x = pw[0]; bm.y = pw[1];         // B[K=c, N=e0+col]
            acc = __builtin_amdgcn_wmma_f32_16x16x4_f32(
                false, a, false, bm, (short)0, acc, false, false);
        }
        const float bias = proj_b[e0 + col];
#pragma unroll
        for (int j = 0; j < 8; ++j) {
            const int row = j + hi * 8;
            zel[row][e0 + col] = acc[j] + bias;
        }
    }
    __syncthreads();

    // ---- per-pixel squared norms of z_e
    if (t < TILE) {
        float s = 0.f;
#pragma unroll 8
        for (int e = 0; e < EDIM; ++e) { float v = zel[t][e]; s += v * v; }
        rownorm[t] = s;
    }
    __syncthreads();

    // ---- GEMM2: x[16 x 512] = 2*z_e.e_k - ||z_e||^2 - ||e_k||^2
    // wave w covers codes [128w, 128w+128)
    {
        v2f afrag[16];
#pragma unroll
        for (int s = 0; s < 16; ++s) {                  // A[M=col, K=4s+kbase..]
            afrag[s].x = zel[col][s * 4 + kbase];
            afrag[s].y = zel[col][s * 4 + kbase + 1];
        }
#pragma unroll
        for (int kt = 0; kt < 8; ++kt) {
            const int k0 = wave * 128 + kt * 16;
            const float* eb = embed_w + (size_t)(k0 + col) * EDIM;
            v8f acc = {};
#pragma unroll
            for (int s = 0; s < 16; ++s) {
                v2f bm;                                  // B[K=e, N=k0+col]
                bm.x = eb[s * 4 + kbase];
                bm.y = eb[s * 4 + kbase + 1];
                acc = __builtin_amdgcn_wmma_f32_16x16x4_f32(
                    false, afrag[s], false, bm, (short)0, acc, false, false);
            }
            const float cn = codenorm[k0 + col];
#pragma unroll
            for (int j = 0; j < 8; ++j) {
                const int row = j + hi * 8;
                xs[row][k0 + col] = 2.f * acc[j] - rownorm[row] - cn;
            }
        }
    }
    __syncthreads();

    // ---- per-row argmax (first occurrence) over 512 codes: 8 threads/row
    {
        const int r  = t >> 3;
        const int c0 = (t & 7) * 64;
        float best = xs[r][c0];
        int   bidx = c0;
        for (int i = 1; i < 64; ++i) {
            const float v = xs[r][c0 + i];
            if (v > best) { best = v; bidx = c0 + i; }   // strict > keeps first max
        }
        pval[t] = best; pidx[t] = bidx;
    }
    __syncthreads();
    if (t < TILE) {
        float best = pval[t * 8];
        int   bidx = pidx[t * 8];
        for (int c = 1; c < 8; ++c) {                    // ordered chunks -> first max
            const float v = pval[t * 8 + c];
            if (v > best) { best = v; bidx = pidx[t * 8 + c]; }
        }
        rowmax[t] = best; rowbest[t] = bidx;
    }
    __syncthreads();

    // ---- logsumexp per row
    {
        const int r  = t >> 3;
        const int c0 = (t & 7) * 64;
        const float mx = rowmax[r];
        float s = 0.f;
        for (int i = 0; i < 64; ++i) s += expf(xs[r][c0 + i] - mx);
        pval[t] = s;
    }
    __syncthreads();
    if (t < TILE) {
        float s = 0.f;
        for (int c = 0; c < 8; ++c) s += pval[t * 8 + c];
        rowlse[t] = rowmax[t] + logf(s);
    }
    __syncthreads();

    // ---- log_priors: [B,K,H,W]; 16 threads (pixel-major) x 8 k per pass
    {
        float* lp = out_lp + (size_t)b * KCODES * HWPIX + hw0;
        const int mm = t & 15;
        const float l = rowlse[mm];
        for (int kk = 0; kk < KCODES / 8; ++kk) {
            const int k = kk * 8 + (t >> 4);
            lp[(size_t)k * HWPIX + mm] = xs[mm][k] - l;
        }
    }

    // ---- indices
    if (t < TILE) out_ind[n0 + t] = (float)rowbest[t];

    // ---- z_q gather + diff partial
    float dsum = 0.f;
    {
        float* zq = out_zq + (size_t)b * EDIM * HWPIX + hw0;
        const int mm = t & 15;
        const float* ev = embed_w + (size_t)rowbest[mm] * EDIM;
        for (int ec = 0; ec < EDIM / 8; ++ec) {
            const int e = ec * 8 + (t >> 4);
            const float v = ev[e];
            zq[(size_t)e * HWPIX + mm] = v;
            const float d = v - zel[mm][e];
            dsum += d * d;
        }
    }
    __syncthreads();
    pval[t] = dsum;
    __syncthreads();
    if (t == 0) {
        float s = 0.f;
        for (int i = 0; i < 128; ++i) s += pval[i];
        diff_part[wg] = s;                 // deterministic: no atomics
    }
}

// ---------------------------------------------------------------------------
// Kernel 2: fixed-order reduction of diff partials -> scalar
// ---------------------------------------------------------------------------
__global__ __launch_bounds__(256) void vq_diff_reduce_kernel(const float* __restrict__ part,
                                                             float* __restrict__ out_diff) {
    __shared__ float sh[256];
    const int t = threadIdx.x;
    float s = 0.f;
    for (int i = t; i < NTILES; i += 256) s += part[i];
    sh[t] = s;
    __syncthreads();
    if (t == 0) {
        float tot = 0.f;
        for (int i = 0; i < 256; ++i) tot += sh[i];
        // diff = 10 * (0.25*mse + mse) = 12.5 * mean over N*E elements
        *out_diff = tot * (12.5f / ((float)NPIX * (float)EDIM));
    }
}

// ---------------------------------------------------------------------------
extern "C" void kernel_launch(void* const* d_in, const int* in_sizes, int n_in,
                              void* d_out, int out_size, void* d_ws, size_t ws_size,
                              hipStream_t stream) {
    const float* z       = (const float*)d_in[0];
    const float* proj_w  = (const float*)d_in[1];
    const float* proj_b  = (const float*)d_in[2];
    const float* embed_w = (const float*)d_in[3];

    float* out = (float*)d_out;
    // outputs concatenated in return order: z_q, diff, ind, log_priors
    const size_t zq_off   = 0;
    const size_t diff_off = (size_t)BATCH * EDIM * HWPIX;     // 4,194,304
    const size_t ind_off  = diff_off + 1;
    const size_t lp_off   = ind_off + (size_t)BATCH * HWPIX;  // 4,259,841

    float* ws_codenorm = (float*)d_ws;              // 512 floats
    float* ws_diffpart = ws_codenorm + KCODES;      // 4096 floats

    vq_prep_kernel<<<1, 512, 0, stream>>>(embed_w, ws_codenorm);

    vq_main_kernel<<<NTILES, 128, 0, stream>>>(
        z, proj_w, proj_b, embed_w, ws_codenorm,
        out + zq_off, out + ind_off, out + lp_off, ws_diffpart);

    vq_diff_reduce_kernel<<<1, 256, 0, stream>>>(ws_diffpart, out + diff_off);
}